// LeadLagSignature_48361331753748
// MI455X (gfx1250) — compile-verified
//
#include <hip/hip_runtime.h>
#include <hip/hip_bf16.h>

// ---------------------------------------------------------------------------
// LeadLagSignature on gfx1250 (MI455X).
//
// Per timestep t: patch = zero-padded rows x[t-15..t] (16 x 8), lead-lag path
// increments e_{2m} = (delta_m, 0), e_{2m+1} = (0, delta_m), delta_m =
// patch[m+1]-patch[m], m = 0..14 (30 increments).
//
//   S1      = sum e_k                                    (16)
//   S2      = sum_k  Ptil_k (x) e_k                      (16x16)   [midpoint]
//   S3 tile_b[a,c] = sum_m (d_m[b] Ptil_{2m}[a]) * (S1[c]-Ptil_{2m}[c])
//                   (b<8: even segments; b>=8: odd segments)
//   plus diagonal-cube correction  -(1/12) sum_m d_m (x) d_m (x) d_m
//   on the lead^3 / lag^3 blocks.
//
// All contractions run on V_WMMA_F32_16X16X4_F32 (true fp32 matrix pipe):
// 8 (S2) + 64 (tile_b) + 16 (cube) = 88 WMMAs per timestep.
// One wave32 per timestep, 4 waves per block, private LDS arena per wave.
// ---------------------------------------------------------------------------

typedef float v2f __attribute__((ext_vector_type(2)));
typedef float v8f __attribute__((ext_vector_type(8)));

__device__ __forceinline__ v8f wmma4(v2f A, v2f B, v8f C) {
  // (neg_a, A, neg_b, B, c_mod, C, reuse_a, reuse_b)
  return __builtin_amdgcn_wmma_f32_16x16x4_f32(false, A, false, B,
                                               (short)0, C, false, false);
}

struct WaveLds {
  float AP2[16][34];  // Ptil^T  [a][k], k=0..29, rows 30,31 zero (S2 A-op)
  float BE[32][16];   // e       [k][b], rows 30,31 zero          (S2 B-op)
  float APe[16][18];  // Ptil_{2m}^T   [a][m], m=0..14, m=15 zero
  float APo[16][18];  // Ptil_{2m+1}^T [a][m]
  float DT[8][18];    // delta^T [ch][m], m=15..17 zero
  float Be[16][16];   // (S1 - Ptil_{2m})   [m][c]
  float Bo[16][16];   // (S1 - Ptil_{2m+1}) [m][c]
  float BG[16][16];   // -delta_m[c]/12 for c<8 else 0  [m][c]
  float G3[512];      // -(1/12) sum_m d[A]d[B]d[C], 8x8x8
};

__global__ __launch_bounds__(128) void
sig_leadlag_kernel(const float* __restrict__ x,
                   float* __restrict__ out1,
                   float* __restrict__ out2,
                   float* __restrict__ out3,
                   int seq_len) {
  __shared__ WaveLds lds[4];

  const int lane = threadIdx.x & 31;
  const int wave = threadIdx.x >> 5;
  const int t    = blockIdx.x * 4 + wave;
  if (t >= seq_len) return;          // uniform per wave: EXEC stays all-ones

  WaveLds& W    = lds[wave];
  const int a   = lane & 15;         // channel / matrix row (M = lane%16)
  const int ch  = a & 7;             // source column of x
  const int half = a >> 3;           // 0 = lead block, 1 = lag block
  const int hi  = lane >> 4;         // lane half
  const int koff = hi ? 2 : 0;       // A/B K-assignment per ISA layout
  const int col = a;                 // N / C column for B and C/D operands

  __builtin_prefetch(x + (size_t)t * 8, 0, 1);

  // ---- load this timestep's patch column (zero padding for t<15) ----------
  float pcol[16];
#pragma unroll
  for (int j = 0; j < 16; ++j) {
    int row = t - 15 + j;
    pcol[j] = (row >= 0) ? x[(size_t)row * 8 + ch] : 0.0f;
  }
  float dlt[15];
#pragma unroll
  for (int m = 0; m < 15; ++m) dlt[m] = pcol[m + 1] - pcol[m];
  const float S1 = pcol[15] - pcol[0];

  // ---- stage operands in this wave's private LDS arena --------------------
  if (lane < 16) {
    float P = 0.0f;
#pragma unroll
    for (int k = 0; k < 30; ++k) {
      const int m  = k >> 1;
      const float e  = ((k & 1) == half) ? dlt[m] : 0.0f;
      const float Pt = P + 0.5f * e;        // midpoint prefix
      const float Bt = S1 - Pt;             // midpoint suffix
      W.AP2[a][k] = Pt;
      W.BE[k][a]  = e;
      if ((k & 1) == 0) { W.APe[a][m] = Pt; W.Be[m][a] = Bt; }
      else              { W.APo[a][m] = Pt; W.Bo[m][a] = Bt; }
      P += e;
    }
    W.AP2[a][30] = 0.0f; W.AP2[a][31] = 0.0f;
    W.APe[a][15] = 0.0f; W.APo[a][15] = 0.0f;
    W.BE[30][a]  = 0.0f; W.BE[31][a]  = 0.0f;
    W.Be[15][a]  = 0.0f; W.Bo[15][a]  = 0.0f;
#pragma unroll
    for (int m = 0; m < 15; ++m) {
      if (half == 0) W.DT[a][m] = dlt[m];
      W.BG[m][a] = (half == 0) ? dlt[m] * (-1.0f / 12.0f) : 0.0f;
    }
    if (half == 0) { W.DT[a][15] = 0.0f; W.DT[a][16] = 0.0f; W.DT[a][17] = 0.0f; }
    W.BG[15][a] = 0.0f;
    out1[(size_t)t * 16 + a] = S1;          // level-1 signature
  }
  __builtin_amdgcn_wave_barrier();          // same-wave DS ops are in-order

  // ---- level 2: S2 = Ptil^T x E, 16x16x32 fp32 GEMM (8 WMMAs) -------------
  {
    v8f acc = {};
#pragma unroll
    for (int c2 = 0; c2 < 8; ++c2) {
      const int k0 = 4 * c2 + koff;
      v2f A = { W.AP2[a][k0], W.AP2[a][k0 + 1] };
      v2f B = { W.BE[k0][col], W.BE[k0 + 1][col] };
      acc = wmma4(A, B, acc);
    }
    float* o2 = out2 + (size_t)t * 256;
#pragma unroll
    for (int v = 0; v < 8; ++v) o2[(v + 8 * hi) * 16 + col] = acc[v];
  }

  // ---- cube correction: G3 = -(1/12) sum_m d(x)d(x)d (16 WMMAs) -----------
#pragma unroll
  for (int g = 0; g < 4; ++g) {
    v8f acc = {};
#pragma unroll
    for (int c4 = 0; c4 < 4; ++c4) {
      const int m0 = 4 * c4 + koff;
      const int R  = 16 * g + a;            // (A,B) row pair, A=R>>3, B=R&7
      const int Ai = R >> 3, Bi = R & 7;
      v2f A = { W.DT[Ai][m0] * W.DT[Bi][m0],
                W.DT[Ai][m0 + 1] * W.DT[Bi][m0 + 1] };
      v2f B = { W.BG[m0][col], W.BG[m0 + 1][col] };
      acc = wmma4(A, B, acc);
    }
    if (col < 8) {
#pragma unroll
      for (int v = 0; v < 8; ++v)
        W.G3[(2 * g + hi) * 64 + v * 8 + col] = acc[v];   // [A][B][C]
    }
  }
  __builtin_amdgcn_wave_barrier();

  // ---- level 3: 16 tiles over middle index b (64 WMMAs) -------------------
  float* o3 = out3 + (size_t)t * 4096;
#pragma unroll
  for (int b = 0; b < 16; ++b) {
    const int bb  = b & 7;
    const int par = b >> 3;                 // 0: even segments, 1: odd
    v8f acc = {};
#pragma unroll
    for (int c4 = 0; c4 < 4; ++c4) {
      const int m0 = 4 * c4 + koff;
      const float p0 = par ? W.APo[a][m0]     : W.APe[a][m0];
      const float p1 = par ? W.APo[a][m0 + 1] : W.APe[a][m0 + 1];
      v2f A = { p0 * W.DT[bb][m0], p1 * W.DT[bb][m0 + 1] };
      v2f B = { par ? W.Bo[m0][col]     : W.Be[m0][col],
                par ? W.Bo[m0 + 1][col] : W.Be[m0 + 1][col] };
      acc = wmma4(A, B, acc);
    }
    // diagonal-cube correction on lead^3 / lag^3 blocks
    if (par == 0) {
      if (lane < 8) {                       // a = v < 8, c = col < 8
#pragma unroll
        for (int v = 0; v < 8; ++v) acc[v] += W.G3[v * 64 + bb * 8 + col];
      }
    } else {
      if (hi == 1 && col >= 8) {            // a = v+8, c = col >= 8
#pragma unroll
        for (int v = 0; v < 8; ++v) acc[v] += W.G3[v * 64 + bb * 8 + (col - 8)];
      }
    }
#pragma unroll
    for (int v = 0; v < 8; ++v)
      o3[(v + 8 * hi) * 256 + b * 16 + col] = acc[v];
  }
}

extern "C" void kernel_launch(void* const* d_in, const int* in_sizes, int n_in,
                              void* d_out, int out_size, void* d_ws, size_t ws_size,
                              hipStream_t stream) {
  (void)n_in; (void)out_size; (void)d_ws; (void)ws_size;
  const float* x = (const float*)d_in[0];
  const int seq = in_sizes[0] / 8;          // SEQ_LEN (DIM = 8)
  float* out  = (float*)d_out;
  float* out1 = out;                                    // [seq][16]
  float* out2 = out + (size_t)seq * 16;                 // [seq][16][16]
  float* out3 = out + (size_t)seq * 16 + (size_t)seq * 256;  // [seq][16][16][16]

  dim3 block(128);                          // 4 wave32 = 4 timesteps / block
  dim3 grid((seq + 3) / 4);
  hipLaunchKernelGGL(sig_leadlag_kernel, grid, block, 0, stream,
                     x, out1, out2, out3, seq);
}